// Resonation_39951785787655
// MI455X (gfx1250) — compile-verified
//
#include <hip/hip_runtime.h>

typedef __attribute__((ext_vector_type(2))) float v2f;
typedef __attribute__((ext_vector_type(8))) float v8f;

#define B_ 8
#define T_ 2048
#define D_ 2048
#define K_ 64
#define R_ (B_ * T_)      // 16384 rows of x (flattened B*T)
#define DC 128            // D-chunk staged in LDS for the GEMM

// ---------------------------------------------------------------------------
// Kernel 1: p = softmax(w, axis=1)  (row-wise over K=64), and wt = w^T
// one thread per row d; reads of w are strided (256B apart) across threads.
// ---------------------------------------------------------------------------
__global__ __launch_bounds__(256) void prep_kernel(const float* __restrict__ w,
                                                   float* __restrict__ p,
                                                   float* __restrict__ wt) {
    int d = blockIdx.x * 256 + threadIdx.x;
    if (d >= D_) return;
    const float* row = w + (size_t)d * K_;
    float m = -__builtin_inff();
#pragma unroll 8
    for (int k = 0; k < K_; ++k) m = fmaxf(m, row[k]);
    float s = 0.0f;
#pragma unroll 8
    for (int k = 0; k < K_; ++k) s += __expf(row[k] - m);
    float inv = 1.0f / s;
#pragma unroll 8
    for (int k = 0; k < K_; ++k) {
        float v = row[k];
        p[(size_t)d * K_ + k]  = __expf(v - m) * inv;
        wt[(size_t)k * D_ + d] = v;   // transpose for coalesced gather later
    }
}

// ---------------------------------------------------------------------------
// Kernel 2: per-column min/max of w over D (64 columns).
// thread k scans column k; each iteration the 64 threads touch one contiguous
// 256B row of w -> fully coalesced.
// ---------------------------------------------------------------------------
__global__ __launch_bounds__(64) void colminmax_kernel(const float* __restrict__ w,
                                                       float* __restrict__ cmin,
                                                       float* __restrict__ cmax) {
    int k = threadIdx.x;
    float mn = w[k], mx = w[k];
    for (int d = 1; d < D_; ++d) {
        float v = w[(size_t)d * K_ + k];
        mn = fminf(mn, v);
        mx = fmaxf(mx, v);
    }
    cmin[k] = mn;
    cmax[k] = mx;
}

// ---------------------------------------------------------------------------
// Kernel 3: scores = x @ p via V_WMMA_F32_16X16X4_F32, then max/argmax over K.
// Block = 256 threads = 8 waves; wave i owns M-tile (16 rows) blockIdx*8+i and
// all 4 N-tiles (K=64 outputs) in 4 v8f accumulators.
// p is staged through LDS in DC=128-row chunks (32 KB); scores are dumped to
// LDS (32 KB) for the cross-lane argmax.
//
// fp32 WMMA fragment layout (ISA 7.12.2):
//   A 16x4 : lanes 0-15 -> M=lane, VGPR0=K0,VGPR1=K1 ; lanes 16-31 -> K2,K3
//   B 4x16 : VGPR0 = rows K0(l0-15)/K2(l16-31), VGPR1 = K1/K3, N = lane%16
//   C/D    : VGPR r = row r (lanes 0-15) / row 8+r (lanes 16-31), N = lane%16
// ---------------------------------------------------------------------------
__global__ __launch_bounds__(256) void score_kernel(const float* __restrict__ x,
                                                    const float* __restrict__ p,
                                                    float* __restrict__ val,
                                                    int* __restrict__ ind) {
    __shared__ float4 sp4[DC * K_ / 4];   // 32 KB: staged chunk of p
    __shared__ float  ss[8 * 16 * K_];    // 32 KB: per-wave 16x64 score tiles
    float* sp = (float*)sp4;

    const int lane  = threadIdx.x & 31;
    const int wave  = threadIdx.x >> 5;
    const int mrow  = lane & 15;          // M (for A) / N (for B,C,D) sub-index
    const int khalf = (lane >> 4) << 1;   // 0 for lanes 0-15, 2 for lanes 16-31
    const int tile  = blockIdx.x * 8 + wave;
    const size_t row0 = (size_t)tile * 16;
    const float* xrow = x + (row0 + (size_t)mrow) * D_;

    v8f acc0 = {0.f,0.f,0.f,0.f,0.f,0.f,0.f,0.f};
    v8f acc1 = acc0, acc2 = acc0, acc3 = acc0;

    for (int dc = 0; dc < D_; dc += DC) {
        // cooperative, coalesced stage of p[dc .. dc+DC) into LDS
        const float4* src = (const float4*)(p + (size_t)dc * K_);
        for (int i = threadIdx.x; i < DC * K_ / 4; i += 256) sp4[i] = src[i];
        __syncthreads();

#pragma unroll 4
        for (int dd = 0; dd < DC; dd += 4) {
            // A fragment: 2 contiguous floats of this row of x
            const float2 av = *(const float2*)(xrow + dc + dd + khalf);
            v2f a; a.x = av.x; a.y = av.y;
            // B fragments for the 4 N-tiles, from LDS
            const float* bb = sp + (dd + khalf) * K_;
            v2f b0; b0.x = bb[mrow];      b0.y = bb[K_ + mrow];
            v2f b1; b1.x = bb[16 + mrow]; b1.y = bb[K_ + 16 + mrow];
            v2f b2; b2.x = bb[32 + mrow]; b2.y = bb[K_ + 32 + mrow];
            v2f b3; b3.x = bb[48 + mrow]; b3.y = bb[K_ + 48 + mrow];
            acc0 = __builtin_amdgcn_wmma_f32_16x16x4_f32(false, a, false, b0, (short)0, acc0, false, false);
            acc1 = __builtin_amdgcn_wmma_f32_16x16x4_f32(false, a, false, b1, (short)0, acc1, false, false);
            acc2 = __builtin_amdgcn_wmma_f32_16x16x4_f32(false, a, false, b2, (short)0, acc2, false, false);
            acc3 = __builtin_amdgcn_wmma_f32_16x16x4_f32(false, a, false, b3, (short)0, acc3, false, false);
        }
        __syncthreads();
    }

    // dump this wave's 16x64 score tile to LDS
    float* myss = ss + wave * 16 * K_;
    const int rhalf = (lane >> 4) * 8;
#pragma unroll
    for (int r = 0; r < 8; ++r) {
        myss[(rhalf + r) * K_ +      mrow] = acc0[r];
        myss[(rhalf + r) * K_ + 16 + mrow] = acc1[r];
        myss[(rhalf + r) * K_ + 32 + mrow] = acc2[r];
        myss[(rhalf + r) * K_ + 48 + mrow] = acc3[r];
    }
    __syncthreads();

    // lanes 0-15: max + argmax over the 64 columns of one row (first-match ties)
    if (lane < 16) {
        const float* srow = myss + lane * K_;
        float best = srow[0];
        int   bi   = 0;
#pragma unroll 8
        for (int c = 1; c < K_; ++c) {
            float v = srow[c];
            if (v > best) { best = v; bi = c; }
        }
        val[row0 + lane] = best;
        ind[row0 + lane] = bi;
    }
}

// ---------------------------------------------------------------------------
// Kernel 4: out[b,t,:] = x[b,t,:] * (1 + Wn[b,t-1,:])   (t==0 -> plain copy)
// Wn[d] = (val*wt[k][d] - mn) / (mx - mn) with
//   mn = min(val*cmin[k], val*cmax[k]), mx = max(...)  (sign-safe)
// One block per row; float4 streaming; gathered column of w is coalesced via wt.
// ---------------------------------------------------------------------------
__global__ __launch_bounds__(256) void apply_kernel(const float* __restrict__ x,
                                                    const float* __restrict__ wt,
                                                    const float* __restrict__ val,
                                                    const int* __restrict__ ind,
                                                    const float* __restrict__ cmin,
                                                    const float* __restrict__ cmax,
                                                    float* __restrict__ out) {
    const size_t i = blockIdx.x;              // flattened (b,t) row, 0..R_-1
    const int    t = (int)(i & (T_ - 1));     // T_ is a power of two
    const float4* xr   = (const float4*)(x + i * D_);
    float4*       orow = (float4*)(out + i * D_);

    if (t == 0) {
        for (int j = threadIdx.x; j < D_ / 4; j += 256) orow[j] = xr[j];
        return;
    }

    const size_t pj = i - 1;                  // stats from previous timestep
    const float  v  = val[pj];
    const int    k  = ind[pj];
    const float  a  = v * cmin[k];
    const float  b  = v * cmax[k];
    const float  mn = fminf(a, b);
    const float  mx = fmaxf(a, b);
    const float  inv = 1.0f / (mx - mn);
    const float4* col = (const float4*)(wt + (size_t)k * D_);

    for (int j = threadIdx.x; j < D_ / 4; j += 256) {
        float4 xv = xr[j];
        float4 cv = col[j];
        float4 o;
        o.x = xv.x * (1.0f + (v * cv.x - mn) * inv);
        o.y = xv.y * (1.0f + (v * cv.y - mn) * inv);
        o.z = xv.z * (1.0f + (v * cv.z - mn) * inv);
        o.w = xv.w * (1.0f + (v * cv.w - mn) * inv);
        orow[j] = o;
    }
}

// ---------------------------------------------------------------------------
extern "C" void kernel_launch(void* const* d_in, const int* in_sizes, int n_in,
                              void* d_out, int out_size, void* d_ws, size_t ws_size,
                              hipStream_t stream) {
    const float* x = (const float*)d_in[0];   // [B,T,D] f32
    const float* w = (const float*)d_in[1];   // [D,K]  f32
    float* out = (float*)d_out;               // [B,T,D] f32

    // workspace layout (~1.2 MB)
    float* p    = (float*)d_ws;               // D_*K_  softmax(w)
    float* wt   = p + (size_t)D_ * K_;        // K_*D_  w transposed
    float* cmin = wt + (size_t)D_ * K_;       // K_
    float* cmax = cmin + K_;                  // K_
    float* val  = cmax + K_;                  // R_
    int*   ind  = (int*)(val + R_);           // R_

    prep_kernel<<<D_ / 256, 256, 0, stream>>>(w, p, wt);
    colminmax_kernel<<<1, K_, 0, stream>>>(w, cmin, cmax);
    score_kernel<<<(R_ / 16) / 8, 256, 0, stream>>>(x, p, val, ind);
    apply_kernel<<<R_, 256, 0, stream>>>(x, wt, val, ind, cmin, cmax, out);
    (void)in_sizes; (void)n_in; (void)out_size; (void)ws_size;
}